// GCN_72593537237727
// MI455X (gfx1250) — compile-verified
//
#include <hip/hip_runtime.h>
#include <hip/hip_bf16.h>

typedef __attribute__((ext_vector_type(2))) float v2f;
typedef __attribute__((ext_vector_type(8))) float v8f;

#define HIDDEN 64

// ---------------------------------------------------------------------------
// Degree / normalization
// ---------------------------------------------------------------------------
__global__ void deg_init_kernel(float* __restrict__ deg, int n) {
    int i = blockIdx.x * blockDim.x + threadIdx.x;
    if (i < n) deg[i] = 1.0f;   // self-loop contributes 1 to every node
}

__global__ void deg_edges_kernel(const int* __restrict__ ei, float* __restrict__ deg, int E) {
    int e = blockIdx.x * blockDim.x + threadIdx.x;
    if (e < E) {
        int dst = ei[E + e];
        atomicAdd(&deg[dst], 1.0f);
    }
}

__global__ void deg_to_dinv_kernel(float* __restrict__ deg, int n) {
    int i = blockIdx.x * blockDim.x + threadIdx.x;
    if (i < n) {
        float d = deg[i];            // >= 1 always (self-loop)
        deg[i] = __frsqrt_rn(d);
    }
}

// ---------------------------------------------------------------------------
// Dense GEMM  H = act(X) @ W   (X: [N,64], W: [64,64] row-major)
// One wave computes a 16x16 tile of H via 16 chained v_wmma_f32_16x16x4_f32.
// Block = 128 threads = 4 waves -> covers 16 rows x 64 cols.
//
// A-fragment (32-bit A 16x4, MxK): lane L holds row M = L%16;
//   VGPR0 = K = kk, VGPR1 = K = kk+1, where kk = (L<16) ? 0 : 2 within the 4-chunk.
// B-fragment (32-bit B 4x16, KxN): symmetric -> lane L holds col N = L%16,
//   VGPR0 = row K = kk, VGPR1 = row K = kk+1.
// C/D (32-bit 16x16): VGPR v holds M = v + 8*(L>=16), N = L%16.
//
// Out-of-range A rows need no masking: A row m only feeds C row m, and
// out-of-range C rows are never stored. Row pointer is merely clamped.
// ---------------------------------------------------------------------------
template <bool RELU>
__global__ void gemm64_wmma_kernel(const float* __restrict__ X,
                                   const float* __restrict__ W,
                                   float* __restrict__ H,
                                   int N) {
    const int lane = threadIdx.x & 31;
    const int wave = threadIdx.x >> 5;         // 0..3 -> which 16-col tile
    const int m0   = blockIdx.x * 16;
    const int n0   = wave * 16;
    const int half = lane >> 4;                // 0 or 1
    const int kk   = half * 2;                 // K sub-offset inside each 4-chunk
    const int mloc = lane & 15;
    const int ncol = n0 + mloc;

    int mrow = m0 + mloc;
    if (mrow >= N) mrow = N - 1;               // clamp only; no zero-masking needed
    const float* xrow = X + (size_t)mrow * HIDDEN;

    // Preload this wave's 64x16 slice of W as 16 B-fragments (32 VGPRs).
    v2f bfrag[16];
#pragma unroll
    for (int s = 0; s < 16; ++s) {
        const int k = 4 * s + kk;
        bfrag[s].x = W[k * HIDDEN + ncol];
        bfrag[s].y = W[(k + 1) * HIDDEN + ncol];
    }

    v8f c = {};
#pragma unroll
    for (int s = 0; s < 16; ++s) {
        const int k = 4 * s + kk;
        float ax = xrow[k];
        float ay = xrow[k + 1];
        if (RELU) {
            ax = fmaxf(ax, 0.0f);
            ay = fmaxf(ay, 0.0f);
        }
        v2f a; a.x = ax; a.y = ay;
        c = __builtin_amdgcn_wmma_f32_16x16x4_f32(
                /*neg_a=*/false, a, /*neg_b=*/false, bfrag[s],
                /*c_mod=*/(short)0, c, /*reuse_a=*/false, /*reuse_b=*/false);
    }

#pragma unroll
    for (int v = 0; v < 8; ++v) {
        const int r = m0 + v + half * 8;
        if (r < N) H[(size_t)r * HIDDEN + ncol] = c[v];
    }
}

// ---------------------------------------------------------------------------
// Aggregation epilogue init:  O[i,:] = H[i,:] * dinv[i]^2 + bias   (self-loop)
// One thread per float4 (16 threads / node).
// ---------------------------------------------------------------------------
__global__ void agg_init_kernel(const float* __restrict__ H,
                                const float* __restrict__ dinv,
                                const float* __restrict__ bias,
                                float* __restrict__ O, int N) {
    int t = blockIdx.x * blockDim.x + threadIdx.x;
    if (t >= N * 16) return;
    int i = t >> 4;
    int c = (t & 15) * 4;
    float d = dinv[i];
    float s = d * d;
    const float4 h = *(const float4*)(H + (size_t)i * HIDDEN + c);
    float4 o;
    o.x = h.x * s + bias[c + 0];
    o.y = h.y * s + bias[c + 1];
    o.z = h.z * s + bias[c + 2];
    o.w = h.w * s + bias[c + 3];
    *(float4*)(O + (size_t)i * HIDDEN + c) = o;
}

// ---------------------------------------------------------------------------
// Edge scatter-add:  O[dst,:] += H[src,:] * dinv[src]*dinv[dst]
// 16 threads per edge, float4 gather + 4x global_atomic_add_f32 (L2 atomics).
// ---------------------------------------------------------------------------
__global__ void agg_edges_kernel(const float* __restrict__ H,
                                 const float* __restrict__ dinv,
                                 const int* __restrict__ ei,
                                 float* __restrict__ O, int E) {
    int t = blockIdx.x * blockDim.x + threadIdx.x;
    if (t >= E * 16) return;
    int e = t >> 4;
    int c = (t & 15) * 4;
    int src = ei[e];
    int dst = ei[E + e];
    float nrm = dinv[src] * dinv[dst];
    const float4 h = *(const float4*)(H + (size_t)src * HIDDEN + c);
    float* o = O + (size_t)dst * HIDDEN + c;
    atomicAdd(o + 0, h.x * nrm);
    atomicAdd(o + 1, h.y * nrm);
    atomicAdd(o + 2, h.z * nrm);
    atomicAdd(o + 3, h.w * nrm);
}

// ---------------------------------------------------------------------------
// Row-wise log_softmax over 64 features: one wave32 per row, 2 elems/lane.
// ---------------------------------------------------------------------------
__global__ void log_softmax_kernel(float* __restrict__ O, int N) {
    const int lane = threadIdx.x & 31;
    const int row  = blockIdx.x * 8 + (threadIdx.x >> 5);
    if (row >= N) return;                        // uniform per wave
    float* p = O + (size_t)row * HIDDEN;
    float v0 = p[lane];
    float v1 = p[lane + 32];
    float m = fmaxf(v0, v1);
#pragma unroll
    for (int off = 16; off > 0; off >>= 1)
        m = fmaxf(m, __shfl_xor(m, off, 32));
    float s = __expf(v0 - m) + __expf(v1 - m);
#pragma unroll
    for (int off = 16; off > 0; off >>= 1)
        s += __shfl_xor(s, off, 32);
    float l = m + __logf(s);
    p[lane]      = v0 - l;
    p[lane + 32] = v1 - l;
}

// ---------------------------------------------------------------------------
// Host-side orchestration (graph-capture safe: launches only)
// ---------------------------------------------------------------------------
extern "C" void kernel_launch(void* const* d_in, const int* in_sizes, int n_in,
                              void* d_out, int out_size, void* d_ws, size_t ws_size,
                              hipStream_t stream) {
    const float* x  = (const float*)d_in[0];
    const int*   ei = (const int*)  d_in[1];
    const float* W1 = (const float*)d_in[2];
    const float* b1 = (const float*)d_in[3];
    const float* W2 = (const float*)d_in[4];
    const float* b2 = (const float*)d_in[5];

    const int N = in_sizes[0] / HIDDEN;
    const int E = in_sizes[1] / 2;

    // Workspace: dinv [N] (padded), bufH [N*64]. d_out serves as the pong buffer.
    float* dinv = (float*)d_ws;
    float* bufH = dinv + (((size_t)N + 63) & ~(size_t)63);
    float* bufT = (float*)d_out;

    const int T = 256;
    const int gN    = (N + T - 1) / T;
    const int gE    = (E + T - 1) / T;
    const int gN16  = (N * 16 + T - 1) / T;
    const int gE16  = (E * 16 + T - 1) / T;
    const int gGemm = (N + 15) / 16;
    const int gRows = (N + 7) / 8;

    // Degree -> dinv (shared by all three convs; deg identical each layer)
    deg_init_kernel   <<<gN, T, 0, stream>>>(dinv, N);
    deg_edges_kernel  <<<gE, T, 0, stream>>>(ei, dinv, E);
    deg_to_dinv_kernel<<<gN, T, 0, stream>>>(dinv, N);

    // Layer 1: h1 = conv(x, W1, b1)            -> bufT (= d_out)
    gemm64_wmma_kernel<false><<<gGemm, 128, 0, stream>>>(x, W1, bufH, N);
    agg_init_kernel  <<<gN16, T, 0, stream>>>(bufH, dinv, b1, bufT, N);
    agg_edges_kernel <<<gE16, T, 0, stream>>>(bufH, dinv, ei, bufT, E);

    // Layer 2: h2 = conv(h1, W1, b1)           -> bufT (ReLU deferred to next load)
    gemm64_wmma_kernel<false><<<gGemm, 128, 0, stream>>>(bufT, W1, bufH, N);
    agg_init_kernel  <<<gN16, T, 0, stream>>>(bufH, dinv, b1, bufT, N);
    agg_edges_kernel <<<gE16, T, 0, stream>>>(bufH, dinv, ei, bufT, E);

    // Layer 3: out = conv(relu(h2), W2, b2)    -> d_out
    gemm64_wmma_kernel<true><<<gGemm, 128, 0, stream>>>(bufT, W2, bufH, N);
    agg_init_kernel  <<<gN16, T, 0, stream>>>(bufH, dinv, b2, (float*)d_out, N);
    agg_edges_kernel <<<gE16, T, 0, stream>>>(bufH, dinv, ei, (float*)d_out, E);

    // log_softmax in-place on d_out
    log_softmax_kernel<<<gRows, T, 0, stream>>>((float*)d_out, N);
}